// MyGlobalAttention_63934883168400
// MI455X (gfx1250) — compile-verified
//
#include <hip/hip_runtime.h>
#include <math.h>

// ---------------------------------------------------------------------------
// Global attention ("general" Luong attention), MI455X / gfx1250.
// Single pass over the 1-GiB memory_bank (online softmax) -> ~46us HBM floor.
// Dense GEMMs use V_WMMA_F32_16X16X4_F32 (full f32, wave32 tile layouts per
// CDNA5 ISA 7.12.2). Streaming pass is VALU f32 (GEMV does not tile onto WMMA)
// with global_prefetch on the next row.
// ---------------------------------------------------------------------------

typedef __attribute__((ext_vector_type(2))) float v2f;
typedef __attribute__((ext_vector_type(8))) float v8f;

#define B_DIM 64
#define S_DIM 4096
#define D_DIM 1024
#define SGROUPS 16                 // S-chunks per batch row
#define ROWS_PER_BLOCK (S_DIM / SGROUPS)   // 256
#define WAVES_PER_BLOCK 8
#define ROWS_PER_WAVE (ROWS_PER_BLOCK / WAVES_PER_BLOCK) // 32

#define NEG_INF (-__builtin_inff())

// ---------------------------------------------------------------------------
// Kernel 1: q = source @ W_in^T   (M=64, N=1024, K=1024), f32 WMMA.
// One wave per 16x16 output tile. A-tile: source rows (M,K); B-tile (K,N):
// B[k][n] = W_in[n][k], i.e. W_in rows read contiguously along K.
// A layout (16x4 f32): lanes 0-15 = M, v0/v1 = K {0,1}; lanes 16-31 = K {2,3}.
// B layout (4x16 f32): lanes 0-15 = N, v0/v1 = K {0,1}; lanes 16-31 = K {2,3}.
// Both reduce to: lane loads float2 at row (base + lane&15), col k0 + 2*(lane>>4).
// ---------------------------------------------------------------------------
__global__ __launch_bounds__(32)
void k_gemm_q(const float* __restrict__ src, const float* __restrict__ Win,
              float* __restrict__ q)
{
    const int lane = threadIdx.x & 31;
    const int half = lane >> 4;
    const int r    = lane & 15;
    const int n0   = (blockIdx.x & 63) << 4;   // 64 N tiles
    const int m0   = (blockIdx.x >> 6) << 4;   // 4  M tiles

    const float* ap = src + (size_t)(m0 + r) * D_DIM + 2 * half;
    const float* bp = Win + (size_t)(n0 + r) * D_DIM + 2 * half;

    v8f acc0 = {}; v8f acc1 = {};
    for (int k0 = 0; k0 < D_DIM; k0 += 8) {
        v2f a0 = *(const v2f*)(ap + k0);
        v2f b0 = *(const v2f*)(bp + k0);
        v2f a1 = *(const v2f*)(ap + k0 + 4);
        v2f b1 = *(const v2f*)(bp + k0 + 4);
        acc0 = __builtin_amdgcn_wmma_f32_16x16x4_f32(false, a0, false, b0,
                                                     (short)0, acc0, false, false);
        acc1 = __builtin_amdgcn_wmma_f32_16x16x4_f32(false, a1, false, b1,
                                                     (short)0, acc1, false, false);
    }
    v8f acc = acc0 + acc1;
    // C/D layout: vgpr v -> M = v + 8*half, N = r
#pragma unroll
    for (int v = 0; v < 8; ++v)
        q[(size_t)(m0 + v + 8 * half) * D_DIM + n0 + r] = acc[v];
}

// ---------------------------------------------------------------------------
// Kernel 2: streaming online-softmax attention over memory_bank.
// Grid (SGROUPS, B). Each wave owns 32 consecutive S-rows; each lane owns the
// D-slice {i*128 + lane*4 + 0..3 | i=0..7} (float4, fully coalesced).
// Row is loaded ONCE into VGPRs, used for score dot AND weighted accumulate.
// ---------------------------------------------------------------------------
__global__ __launch_bounds__(256)
void k_attn_stream(const float* __restrict__ mem, const float* __restrict__ q,
                   const int* __restrict__ lens,
                   float* __restrict__ scores, float* __restrict__ part_c,
                   float* __restrict__ part_ml)
{
    __shared__ float s_m[WAVES_PER_BLOCK];
    __shared__ float s_l[WAVES_PER_BLOCK];
    __shared__ float s_c[D_DIM];

    const int b    = blockIdx.y;
    const int sg   = blockIdx.x;
    const int tid  = threadIdx.x;
    const int lane = tid & 31;
    const int w    = tid >> 5;
    const int len  = lens[b];

    // query slice for this lane
    float4 qv[8];
#pragma unroll
    for (int i = 0; i < 8; ++i)
        qv[i] = *(const float4*)(q + (size_t)b * D_DIM + i * 128 + lane * 4);

    float m = NEG_INF;
    float l = 0.f;
    float4 c[8];
#pragma unroll
    for (int i = 0; i < 8; ++i) c[i] = make_float4(0.f, 0.f, 0.f, 0.f);

    const int j0 = sg * ROWS_PER_BLOCK + w * ROWS_PER_WAVE;
    const float* base = mem + (size_t)b * S_DIM * D_DIM;

    for (int jj = 0; jj < ROWS_PER_WAVE; ++jj) {
        const int j = j0 + jj;
        if (j >= len) break;                       // masked tail: never loaded
        const float* row = base + (size_t)j * D_DIM;
        if (j + 1 < len)                           // prefetch next row (4KB)
            __builtin_prefetch(row + D_DIM + lane * 32, 0, 3);

        float4 h[8];
#pragma unroll
        for (int i = 0; i < 8; ++i)
            h[i] = *(const float4*)(row + i * 128 + lane * 4);

        float p = 0.f;
#pragma unroll
        for (int i = 0; i < 8; ++i)
            p += h[i].x * qv[i].x + h[i].y * qv[i].y +
                 h[i].z * qv[i].z + h[i].w * qv[i].w;
#pragma unroll
        for (int off = 16; off > 0; off >>= 1)     // wave32 butterfly
            p += __shfl_xor(p, off, 32);
        const float s = p;                         // uniform across wave

        if (lane == 0) scores[(size_t)b * S_DIM + j] = s;

        if (s > m) {                               // wave-uniform branch
            const float sc = __expf(m - s);        // first row: exp(-inf)=0
            l *= sc;
#pragma unroll
            for (int i = 0; i < 8; ++i) {
                c[i].x *= sc; c[i].y *= sc; c[i].z *= sc; c[i].w *= sc;
            }
            m = s;
        }
        const float wgt = __expf(s - m);
        l += wgt;
#pragma unroll
        for (int i = 0; i < 8; ++i) {
            c[i].x += wgt * h[i].x; c[i].y += wgt * h[i].y;
            c[i].z += wgt * h[i].z; c[i].w += wgt * h[i].w;
        }
    }

    // --- block reduction over 8 wave partials ---
    if (lane == 0) { s_m[w] = m; s_l[w] = l; }
#pragma unroll
    for (int i = 0; i < 4; ++i) s_c[tid + i * 256] = 0.f;
    __syncthreads();

    float mb = NEG_INF;
#pragma unroll
    for (int g = 0; g < WAVES_PER_BLOCK; ++g) mb = fmaxf(mb, s_m[g]);
    float lb = 0.f;
#pragma unroll
    for (int g = 0; g < WAVES_PER_BLOCK; ++g) {
        const float mg = s_m[g];
        lb += (mg == NEG_INF) ? 0.f : s_l[g] * __expf(mg - mb);
    }

    const float scw = (m == NEG_INF) ? 0.f : __expf(m - mb);
    if (scw != 0.f) {
#pragma unroll
        for (int i = 0; i < 8; ++i) {
            const int d = i * 128 + lane * 4;
            atomicAdd(&s_c[d + 0], scw * c[i].x);  // ds_add_f32
            atomicAdd(&s_c[d + 1], scw * c[i].y);
            atomicAdd(&s_c[d + 2], scw * c[i].z);
            atomicAdd(&s_c[d + 3], scw * c[i].w);
        }
    }
    __syncthreads();

    const int pb = b * SGROUPS + sg;
#pragma unroll
    for (int i = 0; i < 4; ++i)
        part_c[(size_t)pb * D_DIM + tid + i * 256] = s_c[tid + i * 256];
    if (tid == 0) { part_ml[pb * 2] = mb; part_ml[pb * 2 + 1] = lb; }
}

// ---------------------------------------------------------------------------
// Kernel 3: combine the 16 chunk-partials per batch row, normalize by l,
// build cat = [c ; source], save final (m, l).
// ---------------------------------------------------------------------------
__global__ __launch_bounds__(256)
void k_finalize(const float* __restrict__ part_c, const float* __restrict__ part_ml,
                const float* __restrict__ src, float* __restrict__ cat,
                float* __restrict__ final_ml)
{
    const int b   = blockIdx.x;
    const int tid = threadIdx.x;

    float M = NEG_INF;
#pragma unroll
    for (int g = 0; g < SGROUPS; ++g)
        M = fmaxf(M, part_ml[(b * SGROUPS + g) * 2]);

    float L = 0.f;
    float sc[SGROUPS];
#pragma unroll
    for (int g = 0; g < SGROUPS; ++g) {
        const float mg = part_ml[(b * SGROUPS + g) * 2];
        const float lg = part_ml[(b * SGROUPS + g) * 2 + 1];
        const float s  = (mg == NEG_INF) ? 0.f : __expf(mg - M);
        sc[g] = s;
        L += lg * s;
    }

    float4 acc = make_float4(0.f, 0.f, 0.f, 0.f);
#pragma unroll
    for (int g = 0; g < SGROUPS; ++g) {
        if (sc[g] != 0.f) {
            const float4 v =
                *(const float4*)(part_c + (size_t)(b * SGROUPS + g) * D_DIM + tid * 4);
            acc.x += sc[g] * v.x; acc.y += sc[g] * v.y;
            acc.z += sc[g] * v.z; acc.w += sc[g] * v.w;
        }
    }
    const float invL = 1.f / L;                       // len >= 1 so L > 0
    float4 outc = make_float4(acc.x * invL, acc.y * invL, acc.z * invL, acc.w * invL);
    *(float4*)(cat + (size_t)b * 2 * D_DIM + tid * 4) = outc;
    *(float4*)(cat + (size_t)b * 2 * D_DIM + D_DIM + tid * 4) =
        *(const float4*)(src + (size_t)b * D_DIM + tid * 4);
    if (tid == 0) { final_ml[b * 2] = M; final_ml[b * 2 + 1] = L; }
}

// ---------------------------------------------------------------------------
// Kernel 4: align_vectors[b,j] = j<len ? exp(score-m)/l : 0
// ---------------------------------------------------------------------------
__global__ __launch_bounds__(256)
void k_probs(const float* __restrict__ scores, const float* __restrict__ final_ml,
             const int* __restrict__ lens, float* __restrict__ out_align)
{
    const int b = blockIdx.y;
    const int j = blockIdx.x * 256 + threadIdx.x;
    const int len = lens[b];
    const float M = final_ml[b * 2];
    const float L = final_ml[b * 2 + 1];
    float pv = 0.f;
    if (j < len)
        pv = __expf(scores[(size_t)b * S_DIM + j] - M) / L;
    out_align[(size_t)b * S_DIM + j] = pv;
}

// ---------------------------------------------------------------------------
// Kernel 5: attn_h = tanh(cat @ W_out^T)  (M=64, N=1024, K=2048), f32 WMMA.
// ---------------------------------------------------------------------------
__global__ __launch_bounds__(32)
void k_gemm_out(const float* __restrict__ cat, const float* __restrict__ Wout,
                float* __restrict__ out)
{
    const int lane = threadIdx.x & 31;
    const int half = lane >> 4;
    const int r    = lane & 15;
    const int n0   = (blockIdx.x & 63) << 4;
    const int m0   = (blockIdx.x >> 6) << 4;
    const int K    = 2 * D_DIM;

    const float* ap = cat  + (size_t)(m0 + r) * K + 2 * half;
    const float* bp = Wout + (size_t)(n0 + r) * K + 2 * half;

    v8f acc0 = {}; v8f acc1 = {};
    for (int k0 = 0; k0 < K; k0 += 8) {
        v2f a0 = *(const v2f*)(ap + k0);
        v2f b0 = *(const v2f*)(bp + k0);
        v2f a1 = *(const v2f*)(ap + k0 + 4);
        v2f b1 = *(const v2f*)(bp + k0 + 4);
        acc0 = __builtin_amdgcn_wmma_f32_16x16x4_f32(false, a0, false, b0,
                                                     (short)0, acc0, false, false);
        acc1 = __builtin_amdgcn_wmma_f32_16x16x4_f32(false, a1, false, b1,
                                                     (short)0, acc1, false, false);
    }
    v8f acc = acc0 + acc1;
#pragma unroll
    for (int v = 0; v < 8; ++v)
        out[(size_t)(m0 + v + 8 * half) * D_DIM + n0 + r] = tanhf(acc[v]);
}

// ---------------------------------------------------------------------------
extern "C" void kernel_launch(void* const* d_in, const int* in_sizes, int n_in,
                              void* d_out, int out_size, void* d_ws, size_t ws_size,
                              hipStream_t stream)
{
    const float* source      = (const float*)d_in[0];
    const float* memory_bank = (const float*)d_in[1];
    const int*   lens        = (const int*)d_in[2];
    const float* W_in        = (const float*)d_in[3];
    const float* W_out       = (const float*)d_in[4];

    float* out_attn  = (float*)d_out;                    // [64,1024]
    float* out_align = (float*)d_out + B_DIM * D_DIM;    // [64,4096]

    // workspace layout (floats), ~6.04 MB total
    float* ws       = (float*)d_ws;
    float* q        = ws;                                   // 65536
    float* scores   = q + B_DIM * D_DIM;                    // 262144
    float* part_c   = scores + B_DIM * S_DIM;               // 1048576
    float* part_ml  = part_c + (size_t)B_DIM * SGROUPS * D_DIM; // 2048
    float* cat      = part_ml + B_DIM * SGROUPS * 2;        // 131072
    float* final_ml = cat + B_DIM * 2 * D_DIM;              // 128

    k_gemm_q<<<dim3(256), dim3(32), 0, stream>>>(source, W_in, q);
    k_attn_stream<<<dim3(SGROUPS, B_DIM), dim3(256), 0, stream>>>(
        memory_bank, q, lens, scores, part_c, part_ml);
    k_finalize<<<dim3(B_DIM), dim3(256), 0, stream>>>(
        part_c, part_ml, source, cat, final_ml);
    k_probs<<<dim3(S_DIM / 256, B_DIM), dim3(256), 0, stream>>>(
        scores, final_ml, lens, out_align);
    k_gemm_out<<<dim3(256), dim3(32), 0, stream>>>(cat, W_out, out_attn);
}